// Embedding_40200893891216
// MI455X (gfx1250) — compile-verified
//
#include <hip/hip_runtime.h>

// Embedding gather: out[tok, :] = W[:, x[tok]]   (W stored [DIMS, TOKENS] row-major)
// Pure bandwidth problem: 64 MiB coalesced write + ~206 MB scattered read of W.
// Zero FLOPs -> no WMMA; CDNA5 path = async global<->LDS DMA (ASYNCcnt-tracked).
//
// vs round 2: indices fetched via the scalar path (uniform per block) so the
// async loads can use saddr + 32-bit voffset addressing; no LDS index staging,
// no barrier; stores marked non-temporal to protect W's L2 residency
// (W ~= 206 MB vs 192 MB L2, output is write-once).

constexpr int TOKENS  = 50257;
constexpr int DIMS    = 1024;
constexpr int NTOK    = 4 * 4096;   // BATCH * SEQ = 16384
constexpr int TPB     = 8;          // tokens per block
constexpr int THREADS = 256;        // 8 waves (wave32)

#if defined(__HIP_DEVICE_COMPILE__) &&                                   \
    __has_builtin(__builtin_amdgcn_global_load_async_to_lds_b32) &&      \
    __has_builtin(__builtin_amdgcn_global_store_async_from_lds_b128) &&  \
    __has_builtin(__builtin_amdgcn_s_wait_asynccnt)
#define USE_ASYNC 1
#else
#define USE_ASYNC 0
#endif

typedef int v4i __attribute__((ext_vector_type(4)));

typedef __attribute__((address_space(1))) int  g_i32;   // global i32
typedef __attribute__((address_space(3))) int  l_i32;   // LDS i32
typedef __attribute__((address_space(1))) v4i  g_v4i;   // global v4i32
typedef __attribute__((address_space(3))) v4i  l_v4i;   // LDS v4i32

// Cache-policy immediates (gfx12-style CPol: TH in [2:0], scope in [4:3]).
#define CPOL_RT 0
#define CPOL_NT 1

__global__ __launch_bounds__(THREADS) void embed_gather_kernel(
    const int* __restrict__ x, const float* __restrict__ W, float* __restrict__ out)
{
    __shared__ float s_buf[TPB * DIMS];   // 32 KiB staging (of 320 KiB/WGP)

    const int tid  = threadIdx.x;
    const int tok0 = blockIdx.x * TPB;

    // Per-lane byte offset into a W row-block: d = tid*4 + k, stride TOKENS floats.
    // Max = 1023*50257*4 = ~205 MB -> fits unsigned 32-bit (GVS voffset form).
    const unsigned lane_off = (unsigned)(tid * 4) * (unsigned)(TOKENS * 4u);

#if USE_ASYNC
    // Phase 1: gather. Each thread DMA's 4 strided W elements per token into
    // its private 16B LDS slot. tok is uniform (scalar s_load of x), so the
    // global address is saddr(uniform W+tok) + voffset(lane) -> minimal VALU.
    #pragma unroll
    for (int t = 0; t < TPB; ++t) {
        const int tok = x[tok0 + t];                       // uniform -> s_load
        const char* Wtok = (const char*)W + (size_t)tok * 4u;  // uniform base
        #pragma unroll
        for (int k = 0; k < 4; ++k) {
            __builtin_amdgcn_global_load_async_to_lds_b32(
                (g_i32*)(Wtok + lane_off + (unsigned)k * (TOKENS * 4u)),
                (l_i32*)&s_buf[t * DIMS + tid * 4 + k],
                0, CPOL_RT);
        }
    }
    __builtin_amdgcn_s_wait_asynccnt(0);   // per-wave: own LDS bytes ready

    // Phase 2: stream each thread's 16B/token out as a coalesced b128 store,
    // LDS -> global with no VGPR staging; non-temporal to spare L2 for W.
    #pragma unroll
    for (int t = 0; t < TPB; ++t) {
        float* gdst = out + (size_t)(tok0 + t) * DIMS + tid * 4;
        __builtin_amdgcn_global_store_async_from_lds_b128(
            (g_v4i*)gdst, (l_v4i*)&s_buf[t * DIMS + tid * 4],
            0, CPOL_NT);
    }
    __builtin_amdgcn_s_wait_asynccnt(0);   // drain before wave retires
#else
    // Fallback: VGPR-staged gather; 4 scalar loads + one b128 store per token.
    #pragma unroll
    for (int t = 0; t < TPB; ++t) {
        const int tok = x[tok0 + t];
        const float* src = W + tok + (tid * 4) * TOKENS;
        float4 v;
        v.x = src[0 * TOKENS];
        v.y = src[1 * TOKENS];
        v.z = src[2 * TOKENS];
        v.w = src[3 * TOKENS];
        *(float4*)(out + (size_t)(tok0 + t) * DIMS + tid * 4) = v;
    }
    (void)lane_off;
#endif
}

extern "C" void kernel_launch(void* const* d_in, const int* in_sizes, int n_in,
                              void* d_out, int out_size, void* d_ws, size_t ws_size,
                              hipStream_t stream) {
    (void)in_sizes; (void)n_in; (void)out_size; (void)d_ws; (void)ws_size;
    const int*   x = (const int*)d_in[0];      // [4, 4096] int32 token ids
    const float* W = (const float*)d_in[1];    // [1024, 50257] fp32
    float*     out = (float*)d_out;            // [4, 4096, 1024] fp32

    dim3 grid(NTOK / TPB);   // 2048 blocks
    dim3 block(THREADS);     // 256 threads = 8 wave32
    embed_gather_kernel<<<grid, block, 0, stream>>>(x, W, out);
}